// Model_68324339744901
// MI455X (gfx1250) — compile-verified
//
#include <hip/hip_runtime.h>

#define NNODES 50000
#define NHID 128
#define NOUT 64
#define NLAYERS 3
#define ALPHA 0.2f

typedef __attribute__((ext_vector_type(2))) float v2f;
typedef __attribute__((ext_vector_type(8))) float v8f;
typedef __attribute__((ext_vector_type(4))) unsigned int u32x4;
typedef __attribute__((ext_vector_type(4))) int i32x4;
typedef __attribute__((ext_vector_type(8))) int i32x8;

// ---------------- degree / normalization prep ----------------
__global__ void k_fill1(float* __restrict__ deg, int n) {
  int i = blockIdx.x * blockDim.x + threadIdx.x;
  if (i < n) deg[i] = 1.0f;  // self-loop contribution
}

__global__ void k_degcount(const int* __restrict__ dst, float* __restrict__ deg, int E) {
  int e = blockIdx.x * blockDim.x + threadIdx.x;
  if (e < E) atomicAdd(&deg[dst[e]], 1.0f);
}

__global__ void k_rsqrt(float* __restrict__ d, int n) {
  int i = blockIdx.x * blockDim.x + threadIdx.x;
  if (i < n) d[i] = rsqrtf(d[i]);  // deg >= 1 always (self loops)
}

__global__ void k_zero(float* __restrict__ p, int n) {
  int i = blockIdx.x * blockDim.x + threadIdx.x;
  if (i < n) p[i] = 0.0f;
}

// ---------------- dense GEMM: C[M x NCOLS] = A[M x 128] * W[128 x NCOLS] (+bias) ----------------
// One block = 16-row strip of A staged into LDS by the Tensor Data Mover (wave 0
// issues TENSOR_LOAD_TO_LDS with hardware row padding 128->132 dwords for
// bank-conflict-free fragment reads), 4 waves; wave w owns column tiles
// n0 = (w + 4*j)*16. Matrix math: V_WMMA_F32_16X16X4_F32 (fp32, reference-faithful).
#if !(defined(__gfx1250__) && __has_builtin(__builtin_amdgcn_tensor_load_to_lds))
#ifdef __HIP_DEVICE_COMPILE__
#error "gfx1250 TDM builtin unavailable - tensor_load_to_lds path required"
#endif
#endif

template <int NCOLS, bool HAS_BIAS>
__global__ void __launch_bounds__(128) k_gemm_wmma(const float* __restrict__ A,
                                                   const float* __restrict__ W,
                                                   const float* __restrict__ bias,
                                                   float* __restrict__ C) {
  constexpr int K = 128;
  constexpr int PADF = 132;       // padded LDS row pitch in floats (banks 0,4,8,... per row)
  constexpr int NT = NCOLS / 64;  // column tiles per wave (4 waves per block)
  __shared__ float sA[16 * PADF];

  const int tid  = threadIdx.x;
  const int wave = tid >> 5;
  const int lane = tid & 31;
  const int half = lane >> 4;   // 0: lanes 0-15, 1: lanes 16-31
  const int mrow = lane & 15;
  const int m0   = blockIdx.x * 16;

  const float* Ablk = A + (size_t)m0 * K;  // contiguous 16x128 fp32 tile

#if defined(__gfx1250__) && __has_builtin(__builtin_amdgcn_tensor_load_to_lds)
  if (wave == 0) {
    // ---- Tensor DMA descriptor (ISA 8.3/8.4): 2-D tile, 128 x 16 fp32 ----
    unsigned long long gaddr = (unsigned long long)(const void*)Ablk;
    unsigned ldsOff = (unsigned)(unsigned long long)(void*)sA;
    u32x4 g0;
    g0[0] = 1u;                                            // count=1, user mode
    g0[1] = ldsOff;                                        // lds_addr [63:32]
    g0[2] = (unsigned)(gaddr & 0xFFFFFFFFu);               // global_addr lo
    g0[3] = (unsigned)((gaddr >> 32) & 0x1FFFFFFu)         // global_addr hi (bits 56:32)
          | (2u << 30);                                    // type=2 ("image")
    i32x8 g1;
    g1[0] = (2 << 16)                                      // data_size = 4 bytes
          | (1 << 20)                                      // pad_enable (load only)
          | (6 << 22)                                      // pad_interval: every 128 DWORDs
          | (3 << 25);                                     // pad_amount: 4 DWORDs -> pitch 132
    const unsigned td0 = (unsigned)K;                      // tensor_dim0 = 128 elems
    const unsigned td1 = 16u;                              // tensor_dim1 = 16 rows
    g1[1] = (int)((td0 & 0xFFFFu) << 16);                  // td0[15:0] -> bits[63:48]
    g1[2] = (int)(((td0 >> 16) & 0xFFFFu) | ((td1 & 0xFFFFu) << 16));
    g1[3] = (int)(((td1 >> 16) & 0xFFFFu) | (((unsigned)K & 0xFFFFu) << 16));  // tile_dim0=128
    g1[4] = (int)16u;                                      // tile_dim1=16, tile_dim2=0
    g1[5] = (int)(unsigned)K;                              // tensor_dim0_stride = 128
    g1[6] = 0;                                             // stride0 hi / stride1 lo
    g1[7] = 0;                                             // stride1 hi
    i32x4 gz = {0, 0, 0, 0};                               // groups 2/3 unused (<=2-D)
#if __has_include(<hip/amd_detail/amd_gfx1250_TDM.h>)
    i32x8 gz8 = {0, 0, 0, 0, 0, 0, 0, 0};
    __builtin_amdgcn_tensor_load_to_lds(g0, g1, gz, gz, gz8, 0);
#else
    __builtin_amdgcn_tensor_load_to_lds(g0, g1, gz, gz, 0);
#endif
    __builtin_amdgcn_s_wait_tensorcnt(0);
  }
  __syncthreads();
#endif

  v8f acc[NT];
#pragma unroll
  for (int j = 0; j < NT; ++j)
#pragma unroll
    for (int r = 0; r < 8; ++r) acc[j][r] = 0.0f;

  for (int k0 = 0; k0 < K; k0 += 4) {
    // A fragment (16x4 fp32): lanes 0-15 hold K={k0,k0+1}, lanes 16-31 hold K={k0+2,k0+3}
    v2f a;
    a.x = sA[mrow * PADF + k0 + 2 * half + 0];
    a.y = sA[mrow * PADF + k0 + 2 * half + 1];
#pragma unroll
    for (int j = 0; j < NT; ++j) {
      const int n0 = (wave + 4 * j) * 16;
      // B fragment (4x16 fp32): row striped across lanes within a VGPR
      v2f b;
      b.x = W[(size_t)(k0 + 2 * half + 0) * NCOLS + n0 + mrow];
      b.y = W[(size_t)(k0 + 2 * half + 1) * NCOLS + n0 + mrow];
      acc[j] = __builtin_amdgcn_wmma_f32_16x16x4_f32(false, a, false, b, (short)0,
                                                     acc[j], false, false);
    }
  }

#pragma unroll
  for (int j = 0; j < NT; ++j) {
    const int n0 = (wave + 4 * j) * 16;
    const float bb = HAS_BIAS ? bias[n0 + mrow] : 0.0f;
#pragma unroll
    for (int r = 0; r < 8; ++r) {
      // C/D layout: VGPR r -> rows {r, r+8}; N = lane%16
      C[(size_t)(m0 + r + 8 * half) * NCOLS + n0 + mrow] = acc[j][r] + bb;
    }
  }
}

// ---------------- aggregation: out = dinv[i]^2 * vin[i] + bias (float4) ----------------
__global__ void k_selfinit(const float* __restrict__ dinv, const float* __restrict__ vin,
                           const float* __restrict__ bias, float* __restrict__ vout,
                           int total4) {
  int t = blockIdx.x * blockDim.x + threadIdx.x;
  if (t < total4) {
    int i = t >> 5;   // node (32 float4 per node)
    int c4 = t & 31;  // float4 channel group
    float di = dinv[i];
    float d2 = di * di;
    float4 v = ((const float4*)vin)[t];
    float4 b = bias ? ((const float4*)bias)[c4] : make_float4(0.f, 0.f, 0.f, 0.f);
    float4 o;
    o.x = d2 * v.x + b.x;
    o.y = d2 * v.y + b.y;
    o.z = d2 * v.z + b.z;
    o.w = d2 * v.w + b.w;
    ((float4*)vout)[t] = o;
  }
}

// one thread = (edge, 4 channels): b128 gather + 4 L2 f32 atomics
__global__ void k_scatter(const int* __restrict__ src, const int* __restrict__ dst,
                          const float* __restrict__ dinv, const float* __restrict__ vin,
                          float* __restrict__ vout, int E) {
  int t = blockIdx.x * blockDim.x + threadIdx.x;
  int e = t >> 5;
  int c = (t & 31) * 4;
  if (e < E) {
    int s = src[e];
    int d = dst[e];
    float coef = dinv[s] * dinv[d];
    float4 v = *(const float4*)(vin + (size_t)s * 128 + c);
    float* o = vout + (size_t)d * 128 + c;
    atomicAdd(o + 0, coef * v.x);
    atomicAdd(o + 1, coef * v.y);
    atomicAdd(o + 2, coef * v.z);
    atomicAdd(o + 3, coef * v.w);
  }
}

// ---------------- gating + state update (one wave32 per node) ----------------
__global__ void __launch_bounds__(256) k_combine(const float* __restrict__ hbar,
                                                 const float* __restrict__ xlb,
                                                 const float* __restrict__ aggh,
                                                 const float* __restrict__ Wf,
                                                 const float* __restrict__ bf,
                                                 const float* __restrict__ Wu,
                                                 const float* __restrict__ bu,
                                                 float* __restrict__ h) {
  const int lane = threadIdx.x & 31;
  const int node = blockIdx.x * 8 + (threadIdx.x >> 5);
  const size_t base = (size_t)node * 128;

  float sf = 0.0f, su = 0.0f;
  float hv[4], xv[4];
#pragma unroll
  for (int q = 0; q < 4; ++q) {
    int idx = q * 32 + lane;
    hv[q] = hbar[base + idx];
    xv[q] = xlb[base + idx];
    sf += hv[q] * Wf[idx] + xv[q] * Wf[128 + idx];
    su += hv[q] * Wu[idx] + xv[q] * Wu[128 + idx];
  }
  // wave32 butterfly reduction — every lane ends with the full sum
#pragma unroll
  for (int m = 16; m >= 1; m >>= 1) {
    sf += __shfl_xor(sf, m, 32);
    su += __shfl_xor(su, m, 32);
  }
  const float f = 1.0f + ALPHA * tanhf(sf + bf[0]);
  const float u = 1.0f + ALPHA * tanhf(su + bu[0]);
#pragma unroll
  for (int q = 0; q < 4; ++q) {
    int idx = q * 32 + lane;
    h[base + idx] = f * hv[q] + u * xv[q] + aggh[base + idx];
  }
}

// ---------------- driver ----------------
extern "C" void kernel_launch(void* const* d_in, const int* in_sizes, int n_in,
                              void* d_out, int out_size, void* d_ws, size_t ws_size,
                              hipStream_t stream) {
  const float* x   = (const float*)d_in[0];
  const int*   ei  = (const int*)d_in[1];   // [2, E] flattened
  const float* Wh  = (const float*)d_in[2]; // [3,128,128]
  const float* bh  = (const float*)d_in[3]; // [3,128]
  const float* Wx  = (const float*)d_in[4]; // [3,128,128]
  const float* bx  = (const float*)d_in[5]; // [3,128]
  const float* Wf  = (const float*)d_in[6]; // [256]
  const float* bf  = (const float*)d_in[7]; // [1]
  const float* Wu  = (const float*)d_in[8]; // [256]
  const float* bu  = (const float*)d_in[9]; // [1]
  const float* Wfc = (const float*)d_in[10];// [128,64]
  const float* bfc = (const float*)d_in[11];// [64]
  float* out = (float*)d_out;

  const int N  = NNODES;
  const int E  = in_sizes[1] / 2;
  const int NC = N * NHID;

  // workspace layout (floats)
  float* ws   = (float*)d_ws;
  float* dinv = ws;              // N
  float* h    = ws + 50176;      // N*128 each below
  float* t    = h + NC;
  float* hb   = t + NC;
  float* xlb  = hb + NC;
  float* aggh = xlb + NC;

  const int* src = ei;
  const int* dst = ei + E;

  // GCN normalization: dinv = rsqrt(1 + in-degree)
  k_fill1<<<(N + 255) / 256, 256, 0, stream>>>(dinv, N);
  k_degcount<<<(E + 255) / 256, 256, 0, stream>>>(dst, dinv, E);
  k_rsqrt<<<(N + 255) / 256, 256, 0, stream>>>(dinv, N);
  k_zero<<<(NC + 255) / 256, 256, 0, stream>>>(h, NC);  // h0 = 0

  const int gGemm = N / 16;  // 50000 % 16 == 0: no tails, EXEC all-ones for WMMA
  const int gScat = (int)(((long long)E * 32 + 255) / 256);
  const int gElem = (NC / 4 + 255) / 256;

  for (int l = 0; l < NLAYERS; ++l) {
    // _h = agg(h @ Wh[l]) + bh[l]
    k_gemm_wmma<128, false><<<gGemm, 128, 0, stream>>>(h, Wh + (size_t)l * 128 * 128, nullptr, t);
    k_selfinit<<<gElem, 256, 0, stream>>>(dinv, t, bh + l * 128, hb, NC / 4);
    k_scatter<<<gScat, 256, 0, stream>>>(src, dst, dinv, t, hb, E);

    // xl = agg(x @ Wx[l]) + bx[l]
    k_gemm_wmma<128, false><<<gGemm, 128, 0, stream>>>(x, Wx + (size_t)l * 128 * 128, nullptr, t);
    k_selfinit<<<gElem, 256, 0, stream>>>(dinv, t, bx + l * 128, xlb, NC / 4);
    k_scatter<<<gScat, 256, 0, stream>>>(src, dst, dinv, t, xlb, E);

    // agg(h) (residual conv)
    k_selfinit<<<gElem, 256, 0, stream>>>(dinv, h, nullptr, aggh, NC / 4);
    k_scatter<<<gScat, 256, 0, stream>>>(src, dst, dinv, h, aggh, E);

    // gates + update, writes h in place (agg(h) already consumed h)
    k_combine<<<N / 8, 256, 0, stream>>>(hb, xlb, aggh, Wf, bf, Wu, bu, h);
  }

  // out = h @ Wfc + bfc
  k_gemm_wmma<64, true><<<gGemm, 128, 0, stream>>>(h, Wfc, bfc, out);
}